// Mnn_Conv2d_fft_20229295965015
// MI455X (gfx1250) — compile-verified
//
#include <hip/hip_runtime.h>
#include <hip/hip_bf16.h>
#include <math.h>

typedef __attribute__((ext_vector_type(2))) float v2f;
typedef __attribute__((ext_vector_type(8))) float v8f;

#define N4K 4096
#define KB  32
#define LDP 36   // padded LDS stride: even (8B-aligned b64), mult of 4 (16B fills), conflict-free

// M[a,b] = w[((a>>6)-(b>>6)) & 63][((a&63)-(b&63)) & 63], w in LDS
__device__ __forceinline__ float mgen(const float* wl, int a, int b) {
  int row = ((a >> 6) - (b >> 6)) & 63;
  int col = ((a & 63) - (b & 63)) & 63;
  return wl[(row << 6) | col];
}

// One kernel for both GEMM passes over C[r,c] = sum_k A(r,k) * B(k,c), n=4096.
// pass==1: A = Mgen(row0+m, k)        , B = std[k] * corr[k, c]   -> dst = T1'
// pass==2: A = T1'[r,k] * std[k]      , B = Mgen(col0+c, k) (M^T) -> dst = cov_out
// Block: 128 threads = 4 waves; tile 128(M) x 128(N); each wave: 32 rows x 128 cols
// (2 m-subtiles per wave -> each B fragment feeds 2 WMMAs, halving LDS pressure).
__global__ __launch_bounds__(128)
void gemm_mxm(const float* __restrict__ src, const float* __restrict__ stdv,
              const float* __restrict__ wg, float* __restrict__ dst, int pass)
{
  __shared__ float wl[64 * 64];
  __shared__ float Ap[128 * LDP];   // [m 0..127][k 0..31]
  __shared__ float Bp[128 * LDP];   // [c 0..127][k 0..31] (transposed for b64 reads)

  const int tid  = threadIdx.x;
  const int row0 = blockIdx.y * 128;
  const int col0 = blockIdx.x * 128;

  for (int j = tid; j < 4096; j += 128) wl[j] = wg[j];

  v8f acc0[8] = {};   // m-subtile 0 (rows wm*32 + 0..15)
  v8f acc1[8] = {};   // m-subtile 1 (rows wm*32 + 16..31)

  const int wm    = tid >> 5;      // wave id 0..3 -> M rows wm*32 .. wm*32+31
  const int ln    = tid & 31;
  const int lhalf = (ln >> 4);     // 0: K offset 0/1 ; 1: K offset 2/3
  const int l15   = ln & 15;

  for (int k0 = 0; k0 < N4K; k0 += KB) {
    __syncthreads();   // protects panels across iterations; covers wl load on iter 0

    // ---------------- fill A panel: 128 x 32 ----------------
    if (pass == 1) {
      #pragma unroll
      for (int j = 0; j < 32; ++j) {
        int idx = tid + j * 128;                // 4096 elements
        int m = idx >> 5, k = idx & 31;
        Ap[m * LDP + k] = mgen(wl, row0 + m, k0 + k);
      }
    } else {
      #pragma unroll
      for (int j = 0; j < 8; ++j) {
        int slot = tid + j * 128;               // 1024 float4 slots
        int m = slot >> 3, k4 = (slot & 7) << 2;
        float4 v  = *(const float4*)(src  + (size_t)(row0 + m) * N4K + k0 + k4);
        float4 s4 = *(const float4*)(stdv + k0 + k4);
        v.x *= s4.x; v.y *= s4.y; v.z *= s4.z; v.w *= s4.w;
        *(float4*)&Ap[m * LDP + k4] = v;        // 16B aligned: LDP%4==0, k4%4==0
      }
    }

    // ---------------- fill B panel: 128 x 32 ----------------
    if (pass == 1) {
      #pragma unroll
      for (int j = 0; j < 8; ++j) {
        int slot = tid + j * 128;               // 1024 float4 slots
        int k = slot >> 5, c4 = (slot & 31) << 2;
        float4 v = *(const float4*)(src + (size_t)(k0 + k) * N4K + col0 + c4);
        float s = stdv[k0 + k];
        Bp[(c4 + 0) * LDP + k] = v.x * s;
        Bp[(c4 + 1) * LDP + k] = v.y * s;
        Bp[(c4 + 2) * LDP + k] = v.z * s;
        Bp[(c4 + 3) * LDP + k] = v.w * s;
      }
    } else {
      #pragma unroll
      for (int j = 0; j < 32; ++j) {
        int idx = tid + j * 128;                // 4096 elements
        int c = idx >> 5, k = idx & 31;
        Bp[c * LDP + k] = mgen(wl, col0 + c, k0 + k);   // M^T(k,c) = M[c',k]
      }
    }

    __syncthreads();

    // ---------------- compute: 8 WMMA k-steps of 4, 16 WMMA each ----------------
    #pragma unroll
    for (int kk = 0; kk < KB; kk += 4) {
      const int kl = kk + (lhalf << 1);   // lanes 0-15: K0/K1 ; lanes 16-31: K2/K3
      v2f a0 = *(const v2f*)&Ap[(wm * 32 +      l15) * LDP + kl];
      v2f a1 = *(const v2f*)&Ap[(wm * 32 + 16 + l15) * LDP + kl];
      #pragma unroll
      for (int s = 0; s < 8; ++s) {
        v2f bvec = *(const v2f*)&Bp[(s * 16 + l15) * LDP + kl];
        acc0[s] = __builtin_amdgcn_wmma_f32_16x16x4_f32(
            false, a0, false, bvec, (short)0, acc0[s], false, false);
        acc1[s] = __builtin_amdgcn_wmma_f32_16x16x4_f32(
            false, a1, false, bvec, (short)0, acc1[s], false, false);
      }
    }
  }

  // ---------------- store: VGPR p -> M=p (lanes 0-15) / M=p+8 (lanes 16-31) ----------------
  const int r_base0 = row0 + wm * 32 + (lhalf ? 8 : 0);
  const int r_base1 = r_base0 + 16;
  #pragma unroll
  for (int s = 0; s < 8; ++s) {
    int c = col0 + s * 16 + l15;
    #pragma unroll
    for (int p = 0; p < 8; ++p) {
      dst[(size_t)(r_base0 + p) * N4K + c] = acc0[s][p];
      dst[(size_t)(r_base1 + p) * N4K + c] = acc1[s][p];
    }
  }
}

// mean_out[b] = M * mean[b]  (circular 2D conv), tiny cost
__global__ __launch_bounds__(256)
void mean_conv(const float* __restrict__ mean_in, const float* __restrict__ wg,
               float* __restrict__ mean_out)
{
  __shared__ float wl[4096];
  __shared__ float ml[4096];
  const int b = blockIdx.y, tid = threadIdx.x;
  for (int j = tid; j < 4096; j += 256) {
    wl[j] = wg[j];
    ml[j] = mean_in[b * 4096 + j];
  }
  __syncthreads();
  const int r = blockIdx.x * 256 + tid;
  const int ri = r >> 6, rj = r & 63;
  float acc = 0.f;
  for (int ki = 0; ki < 64; ++ki) {
    const float* wrow = &wl[(((ri - ki) & 63)) << 6];
    const float* mrow = &ml[ki << 6];
    #pragma unroll 8
    for (int kj = 0; kj < 64; ++kj)
      acc += wrow[(rj - kj) & 63] * mrow[kj];
  }
  mean_out[b * 4096 + r] = acc;
}

__global__ void std_diag(const float* __restrict__ cov, float* __restrict__ std_out)
{
  int i = blockIdx.x * 256 + threadIdx.x;    // 0..8191
  int b = i >> 12, r = i & 4095;
  float v = cov[(size_t)b * N4K * N4K + (size_t)r * (N4K + 1)];
  std_out[i] = sqrtf(fmaxf(v, 1e-12f));
}

__global__ void normalize_corr(float* __restrict__ corr, const float* __restrict__ std_o)
{
  size_t e = ((size_t)blockIdx.x * 256 + threadIdx.x) * 4;   // 2^25 elems total, no tail
  size_t b  = e >> 24;               // 4096*4096 = 2^24
  size_t rc = e & 0xFFFFFFu;
  int r = (int)(rc >> 12);
  int c = (int)(rc & 4095);
  float sr = std_o[(b << 12) | r];
  const float* sc = &std_o[(b << 12) | c];
  float4 v = *(float4*)&corr[e];
  v.x = v.x / (sr * sc[0]);
  v.y = v.y / (sr * sc[1]);
  v.z = v.z / (sr * sc[2]);
  v.w = v.w / (sr * sc[3]);
  *(float4*)&corr[e] = v;
}

extern "C" void kernel_launch(void* const* d_in, const int* in_sizes, int n_in,
                              void* d_out, int out_size, void* d_ws, size_t ws_size,
                              hipStream_t stream)
{
  const float* mean_in = (const float*)d_in[0];   // (2,64,64)
  const float* std_in  = (const float*)d_in[1];   // (2,64,64)
  const float* corr_in = (const float*)d_in[2];   // (2,4096,4096)
  const float* weight  = (const float*)d_in[3];   // (64,64)

  float* out      = (float*)d_out;
  float* mean_out = out;                    // 2*4096
  float* std_out  = out + 2 * 4096;         // 2*4096
  float* corr_out = out + 4 * 4096;         // 2*4096*4096
  float* T1       = (float*)d_ws;           // 4096*4096 floats scratch

  dim3 gblk(128);
  dim3 ggrid(N4K / 128, N4K / 128);         // (32, 32)

  mean_conv<<<dim3(16, 2), 256, 0, stream>>>(mean_in, weight, mean_out);

  for (int b = 0; b < 2; ++b) {
    const float* corr_b = corr_in + (size_t)b * N4K * N4K;
    const float* std_b  = std_in + b * 4096;
    float* cov_b        = corr_out + (size_t)b * N4K * N4K;
    // T1' = M * (D * corr)
    gemm_mxm<<<ggrid, gblk, 0, stream>>>(corr_b, std_b, weight, T1, 1);
    // cov = (T1' * D) * M^T
    gemm_mxm<<<ggrid, gblk, 0, stream>>>(T1, std_b, weight, cov_b, 2);
  }

  std_diag<<<32, 256, 0, stream>>>(corr_out, std_out);
  normalize_corr<<<32768, 256, 0, stream>>>(corr_out, std_out);
}